// SkillParameters_88974542504075
// MI455X (gfx1250) — compile-verified
//
#include <hip/hip_runtime.h>
#include <hip/hip_bf16.h>

typedef __attribute__((ext_vector_type(2))) float v2f;
typedef __attribute__((ext_vector_type(4))) float v4f;
typedef __attribute__((ext_vector_type(8))) float v8f;

#define N_PLAYERS 256
#define T_STEPS   1024
#define P_ORDER   4
#define ALPHA_LD  (T_STEPS + P_ORDER)   // 1028 columns in alpha
#define TP        (T_STEPS - P_ORDER)   // 1020

// ws layout (floats): [0] = btl accumulator, [16..16+1023] = Ssum[t], [16+1024..] = Ssq[t]
#define WS_BTL    0
#define WS_SSUM   16
#define WS_SSQ    (16 + 1024)
#define WS_FLOATS (16 + 2048)

// ---------------------------------------------------------------------------
// Zero the small workspace accumulators (harness does not re-poison between
// graph replays, so we must re-init every launch).
// ---------------------------------------------------------------------------
__global__ __launch_bounds__(256) void init_ws(float* __restrict__ ws) {
  int idx = blockIdx.x * blockDim.x + threadIdx.x;
  if (idx < WS_FLOATS) ws[idx] = 0.0f;
}

// ---------------------------------------------------------------------------
// log-BTL term: memory-bound stream of the Z/W upper triangle (~267 MB).
// One block per (i,j) pair; 256 threads x float4 = 1024 timesteps.
// Non-temporal loads on Z/W keep the heavily reused alpha rows cached.
// ---------------------------------------------------------------------------
__device__ __forceinline__ float btl_elem(float z, float w, float si, float sj) {
  float m = fmaxf(si, sj);
  float L = m + __logf(__expf(si - m) + __expf(sj - m));   // logaddexp
  return z * (si - L) + w * (sj - L);
}

__global__ __launch_bounds__(256) void btl_kernel(const float* __restrict__ Z,
                                                  const float* __restrict__ W,
                                                  const float* __restrict__ alpha,
                                                  float* __restrict__ ws) {
  const int i = blockIdx.y;
  const int j = blockIdx.x;
  if (j <= i) return;                       // upper triangle only
  const int tid = threadIdx.x;
  const size_t pair = (size_t)(i * N_PLAYERS + j) * T_STEPS;

  const v4f z  = __builtin_nontemporal_load((const v4f*)(Z + pair) + tid);
  const v4f w  = __builtin_nontemporal_load((const v4f*)(W + pair) + tid);
  const v4f si = *((const v4f*)(alpha + i * ALPHA_LD + P_ORDER) + tid);  // alpha[:, -T:]
  const v4f sj = *((const v4f*)(alpha + j * ALPHA_LD + P_ORDER) + tid);

  float acc = btl_elem(z.x, w.x, si.x, sj.x)
            + btl_elem(z.y, w.y, si.y, sj.y)
            + btl_elem(z.z, w.z, si.z, sj.z)
            + btl_elem(z.w, w.w, si.w, sj.w);

  __shared__ float red[256];
  red[tid] = acc;
  __syncthreads();
  for (int s = 128; s > 0; s >>= 1) {
    if (tid < s) red[tid] += red[tid + s];
    __syncthreads();
  }
  if (tid == 0) atomicAdd(ws + WS_BTL, red[0]);
}

// ---------------------------------------------------------------------------
// AR einsum as f32 WMMA GEMM:
//   summed[n,t] = sum_{k,m} Phi[k,n,m] * alpha[m, k+t]
// A is (256 x 1024) with K = k*256+m, B is (1024 x 1020).
// One wave per 16x16 output tile; 256 steps of v_wmma_f32_16x16x4_f32.
// Per-tile we fold directly into per-t column statistics (Ssum, Ssq) so the
// 256x1020 intermediate never hits memory.
// ---------------------------------------------------------------------------
__global__ __launch_bounds__(32) void ar_gemm_wmma(const float* __restrict__ Phi,
                                                   const float* __restrict__ alpha,
                                                   float* __restrict__ wsSsum,
                                                   float* __restrict__ wsSsq) {
  const int tile = blockIdx.x;      // 16 row tiles x 64 col tiles = 1024
  const int n0 = (tile & 15) * 16;  // output rows  n0..n0+15
  const int t0 = (tile >> 4) * 16;  // output cols  t0..t0+15  (cols >= 1020 masked)
  const int lane = threadIdx.x;
  const int lo = lane & 15;
  const int hi = lane >> 4;         // 0: K slots {0,1}, 1: K slots {2,3}

  const int rowA = n0 + lo;         // A-fragment row (ISA: lanes 0-15 / 16-31 both M=0..15)
  const int colB = t0 + lo;         // B-fragment col

  v8f c = {};
  for (int k = 0; k < 1024; k += 4) {
    const int kk0 = k + 2 * hi;
    const int kk1 = kk0 + 1;
    v2f a, b;
    // A[n, kk] = Phi[(kk>>8)*65536 + n*256 + (kk&255)]
    a.x = Phi[((kk0 >> 8) << 16) + (rowA << 8) + (kk0 & 255)];
    a.y = Phi[((kk1 >> 8) << 16) + (rowA << 8) + (kk1 & 255)];
    // B[kk, t] = alpha[(kk&255)*1028 + (kk>>8) + t]   (k+t <= 1026 < 1028, in-bounds)
    b.x = alpha[(kk0 & 255) * ALPHA_LD + (kk0 >> 8) + colB];
    b.y = alpha[(kk1 & 255) * ALPHA_LD + (kk1 >> 8) + colB];
    c = __builtin_amdgcn_wmma_f32_16x16x4_f32(false, a, false, b, (short)0, c,
                                              false, false);
  }

  // C/D layout: VGPR r holds row M = r + 8*hi, column N = lane&15.
  // Column t = t0+lo lives in lanes {lo, lo+16} -> reduce pair, then lane lo commits.
  float ls = 0.f, lsq = 0.f;
#pragma unroll
  for (int r = 0; r < 8; ++r) {
    float v = c[r];
    ls += v;
    lsq += v * v;
  }
  ls  += __shfl_xor(ls, 16, 32);
  lsq += __shfl_xor(lsq, 16, 32);
  const int t = t0 + lo;
  if (hi == 0 && t < TP) {
    atomicAdd(&wsSsum[t], ls);
    atomicAdd(&wsSsq[t], lsq);
  }
}

// ---------------------------------------------------------------------------
// Finalize:
//   ar_err = sum_t [ N*Ssq[t] + N*Asq[t] - 2*Ssum[t]*Asum[t] ]
// where actual[m,t] = alpha[m, P+t]. Also copies the BTL accumulator out.
// ---------------------------------------------------------------------------
__global__ __launch_bounds__(256) void finalize(const float* __restrict__ alpha,
                                                const float* __restrict__ ws,
                                                float* __restrict__ out) {
  const int tid = threadIdx.x;
  float acc = 0.f;
  for (int t = tid; t < TP; t += 256) {
    float asum = 0.f, asq = 0.f;
    for (int m = 0; m < N_PLAYERS; ++m) {
      float v = alpha[m * ALPHA_LD + P_ORDER + t];
      asum += v;
      asq += v * v;
    }
    const float ssum = ws[WS_SSUM + t];
    const float ssq  = ws[WS_SSQ + t];
    acc += (float)N_PLAYERS * ssq + (float)N_PLAYERS * asq - 2.f * ssum * asum;
  }
  __shared__ float red[256];
  red[tid] = acc;
  __syncthreads();
  for (int s = 128; s > 0; s >>= 1) {
    if (tid < s) red[tid] += red[tid + s];
    __syncthreads();
  }
  if (tid == 0) {
    out[0] = ws[WS_BTL];  // log_btl
    out[1] = red[0];      // ar_err
  }
}

// ---------------------------------------------------------------------------
extern "C" void kernel_launch(void* const* d_in, const int* in_sizes, int n_in,
                              void* d_out, int out_size, void* d_ws, size_t ws_size,
                              hipStream_t stream) {
  const float* Z     = (const float*)d_in[0];
  const float* W     = (const float*)d_in[1];
  const float* Phi   = (const float*)d_in[2];
  const float* alpha = (const float*)d_in[3];
  float* ws  = (float*)d_ws;
  float* out = (float*)d_out;

  init_ws<<<(WS_FLOATS + 255) / 256, 256, 0, stream>>>(ws);
  btl_kernel<<<dim3(N_PLAYERS, N_PLAYERS), 256, 0, stream>>>(Z, W, alpha, ws);
  ar_gemm_wmma<<<16 * 64, 32, 0, stream>>>(Phi, alpha, ws + WS_SSUM, ws + WS_SSQ);
  finalize<<<1, 256, 0, stream>>>(alpha, ws, out);
}